// GumbelSubsetOperator_1400159339070
// MI455X (gfx1250) — compile-verified
//
#include <hip/hip_runtime.h>
#include <cstdint>
#include <cstddef>

// Problem constants (match reference)
#define B_ROWS   2048
#define N_COLS   8192
#define K_STEPS  64
#define TPB      1024
#define EPT      (N_COLS / TPB)   // 8 elements per thread
#define VEC      4
#define F4_PER_T (EPT / VEC)      // 2 float4 per thread
#define EPS_K    1e-10f

// The async builtin wants int4 pointers qualified with explicit address spaces:
//   arg0: __attribute__((address_space(1))) int4*   (global)
//   arg1: __attribute__((address_space(3))) int4*   (LDS)
typedef int v4i __attribute__((ext_vector_type(4)));
typedef __attribute__((address_space(1))) v4i as1_v4i;
typedef __attribute__((address_space(3))) v4i as3_v4i;
typedef __attribute__((address_space(1))) void as1_void;
typedef __attribute__((address_space(3))) void as3_void;

// ---- gfx1250 async global->LDS staging (ASYNCcnt path) ------------------
__device__ __forceinline__ void stage_b128(const float* gsrc, float* ldst) {
#if defined(__HIP_DEVICE_COMPILE__) && __has_builtin(__builtin_amdgcn_global_load_async_to_lds_b128)
    __builtin_amdgcn_global_load_async_to_lds_b128(
        (as1_v4i*)(as1_void*)(gsrc),
        (as3_v4i*)(as3_void*)(ldst),
        /*imm offset*/0, /*cpol*/0);
#else
    // Fallback: synchronous copy (still correct, just not async)
    *(float4*)ldst = *(const float4*)gsrc;
#endif
}

__device__ __forceinline__ void stage_wait() {
#if defined(__HIP_DEVICE_COMPILE__)
#if __has_builtin(__builtin_amdgcn_s_wait_asynccnt)
    __builtin_amdgcn_s_wait_asynccnt(0);
#else
    asm volatile("s_wait_asynccnt 0" ::: "memory");
#endif
#endif
}

// ---- wave32 butterfly reductions via ds_swizzle (no bounds/cndmask) -----
// Group-of-32 swizzle encoding: offset[15]=0, xor_mask=[14:10], or=[9:5], and=[4:0]
// SWAPX{1,2,4,8,16} = (xor << 10) | 0x1F
template <int PATTERN>
__device__ __forceinline__ float swz_xor(float v) {
#if defined(__HIP_DEVICE_COMPILE__) && __has_builtin(__builtin_amdgcn_ds_swizzle)
    return __int_as_float(__builtin_amdgcn_ds_swizzle(__float_as_int(v), PATTERN));
#else
    return __shfl_xor(v, (PATTERN >> 10) & 0x1F, 32);
#endif
}

__device__ __forceinline__ float wred_max(float v) {
    v = fmaxf(v, swz_xor<(1  << 10) | 0x1F>(v));
    v = fmaxf(v, swz_xor<(2  << 10) | 0x1F>(v));
    v = fmaxf(v, swz_xor<(4  << 10) | 0x1F>(v));
    v = fmaxf(v, swz_xor<(8  << 10) | 0x1F>(v));
    v = fmaxf(v, swz_xor<(16 << 10) | 0x1F>(v));
    return v;
}

__device__ __forceinline__ float wred_sum(float v) {
    v += swz_xor<(1  << 10) | 0x1F>(v);
    v += swz_xor<(2  << 10) | 0x1F>(v);
    v += swz_xor<(4  << 10) | 0x1F>(v);
    v += swz_xor<(8  << 10) | 0x1F>(v);
    v += swz_xor<(16 << 10) | 0x1F>(v);
    return v;
}

// -------------------------------------------------------------------------
// One workgroup per row. Row state lives in VGPRs for all K iterations;
// HBM traffic is exactly: read scores+g once, write khot once.
// -------------------------------------------------------------------------
extern "C" __global__ void __launch_bounds__(TPB)
gumbel_subset_khot_kernel(const float* __restrict__ scores,
                          const float* __restrict__ g,
                          float* __restrict__ out)
{
    __shared__ __align__(16) float ldsS[N_COLS];  // staged scores row (32 KB)
    __shared__ float redMax[32];
    __shared__ float redSum[32];

    const int row  = blockIdx.x;
    const int t    = (int)threadIdx.x;
    const int lane = t & 31;
    const int wid  = t >> 5;

    const float* srow = scores + (size_t)row * N_COLS;
    const float* grow = g      + (size_t)row * N_COLS;

    // Issue async global->LDS copies for the scores row (128b per lane).
#pragma unroll
    for (int c = 0; c < F4_PER_T; ++c) {
        const int fi = (t + c * TPB) * VEC;   // 16B-aligned float index
        stage_b128(srow + fi, &ldsS[fi]);
    }

    // Overlap: pull the Gumbel noise straight into registers with b128 loads.
    float s[EPT], p[EPT], kh[EPT];
#pragma unroll
    for (int c = 0; c < F4_PER_T; ++c) {
        const float4 v = *(const float4*)(grow + t * EPT + c * VEC);
        s[c * VEC + 0] = v.x;
        s[c * VEC + 1] = v.y;
        s[c * VEC + 2] = v.z;
        s[c * VEC + 3] = v.w;
    }

    stage_wait();        // this wave's async copies landed in LDS
    __syncthreads();     // all waves' copies visible

    // s0 = scores + g ; p = 0 ; khot = 0
#pragma unroll
    for (int c = 0; c < F4_PER_T; ++c) {
        const float4 v = *(const float4*)&ldsS[t * EPT + c * VEC];
        s[c * VEC + 0] += v.x;
        s[c * VEC + 1] += v.y;
        s[c * VEC + 2] += v.z;
        s[c * VEC + 3] += v.w;
    }
#pragma unroll
    for (int j = 0; j < EPT; ++j) { p[j] = 0.0f; kh[j] = 0.0f; }

    // K relaxation steps, fully on-chip
    for (int it = 0; it < K_STEPS; ++it) {
        // s += log(max(1 - p, eps)); track local max
        float lmax = -3.402823466e+38f;
#pragma unroll
        for (int j = 0; j < EPT; ++j) {
            s[j] += __logf(fmaxf(1.0f - p[j], EPS_K));
            lmax  = fmaxf(lmax, s[j]);
        }

        // Block-wide max (wave reduce -> LDS -> wave reduce)
        lmax = wred_max(lmax);
        if (lane == 0) redMax[wid] = lmax;
        __syncthreads();
        const float m = wred_max(redMax[lane]);

        // exp(s - m) and block-wide sum
        float lsum = 0.0f;
#pragma unroll
        for (int j = 0; j < EPT; ++j) {
            p[j]  = __expf(s[j] - m);
            lsum += p[j];
        }
        lsum = wred_sum(lsum);
        if (lane == 0) redSum[wid] = lsum;
        __syncthreads();
        const float z   = wred_sum(redSum[lane]);
        const float inv = 1.0f / z;

        // normalize and accumulate khot
#pragma unroll
        for (int j = 0; j < EPT; ++j) {
            p[j] *= inv;
            kh[j] += p[j];
        }
    }

    // Write khot row once, b128 stores
    float* orow = out + (size_t)row * N_COLS + t * EPT;
#pragma unroll
    for (int c = 0; c < F4_PER_T; ++c) {
        *(float4*)(orow + c * VEC) =
            make_float4(kh[c * VEC + 0], kh[c * VEC + 1],
                        kh[c * VEC + 2], kh[c * VEC + 3]);
    }
}

extern "C" void kernel_launch(void* const* d_in, const int* in_sizes, int n_in,
                              void* d_out, int out_size, void* d_ws, size_t ws_size,
                              hipStream_t stream) {
    (void)n_in; (void)out_size; (void)d_ws; (void)ws_size;
    const float* scores = (const float*)d_in[0];
    const float* g      = (const float*)d_in[1];
    float* out          = (float*)d_out;

    const int rows = in_sizes[0] / N_COLS;   // 2048
    gumbel_subset_khot_kernel<<<dim3(rows), dim3(TPB), 0, stream>>>(scores, g, out);
}